// EMDloss_36344013258961
// MI455X (gfx1250) — compile-verified
//
#include <hip/hip_runtime.h>
#include <math.h>

#define N_BINS 85

typedef __attribute__((ext_vector_type(2))) float v2f;
typedef __attribute__((ext_vector_type(8))) float v8f;

// ---------------------------------------------------------------------------
// Kernel 1: zero the 85-float accumulator in workspace (d_ws is poisoned).
// ---------------------------------------------------------------------------
__global__ __launch_bounds__(128) void emd_zero_ws(float* __restrict__ ws) {
    if (threadIdx.x < N_BINS) ws[threadIdx.x] = 0.0f;
}

// ---------------------------------------------------------------------------
// Load one 4-row group's worth of lane data: tiles 0..4 from pA (cols n+16t,
// rows k={2*khalf, 2*khalf+1}), tile 5 from the clamped pointer p5.
// 12 independent loads -> one clause, no EXEC changes.
// ---------------------------------------------------------------------------
__device__ __forceinline__ void emd_load_group(const float* __restrict__ pA,
                                               const float* __restrict__ p5,
                                               float v[12]) {
#pragma unroll
    for (int t = 0; t < 5; ++t) {
        v[2 * t]     = pA[t * 16];            // k = 2*khalf
        v[2 * t + 1] = pA[N_BINS + t * 16];   // k = 2*khalf + 1
    }
    v[10] = p5[0];
    v[11] = p5[N_BINS];
}

// ---------------------------------------------------------------------------
// 12 subs (tile 5 masked by m5) then 6 WMMA accumulations.
// ---------------------------------------------------------------------------
__device__ __forceinline__ void emd_accum(const float vi[12], const float vg[12],
                                          float m5, v2f ones,
                                          v8f& a0, v8f& a1, v8f& a2,
                                          v8f& a3, v8f& a4, v8f& a5) {
    v2f b;
    b[0] = vi[0] - vg[0];  b[1] = vi[1] - vg[1];
    a0 = __builtin_amdgcn_wmma_f32_16x16x4_f32(false, ones, false, b, (short)0, a0, false, false);
    b[0] = vi[2] - vg[2];  b[1] = vi[3] - vg[3];
    a1 = __builtin_amdgcn_wmma_f32_16x16x4_f32(false, ones, false, b, (short)0, a1, false, false);
    b[0] = vi[4] - vg[4];  b[1] = vi[5] - vg[5];
    a2 = __builtin_amdgcn_wmma_f32_16x16x4_f32(false, ones, false, b, (short)0, a2, false, false);
    b[0] = vi[6] - vg[6];  b[1] = vi[7] - vg[7];
    a3 = __builtin_amdgcn_wmma_f32_16x16x4_f32(false, ones, false, b, (short)0, a3, false, false);
    b[0] = vi[8] - vg[8];  b[1] = vi[9] - vg[9];
    a4 = __builtin_amdgcn_wmma_f32_16x16x4_f32(false, ones, false, b, (short)0, a4, false, false);
    b[0] = (vi[10] - vg[10]) * m5;  b[1] = (vi[11] - vg[11]) * m5;
    a5 = __builtin_amdgcn_wmma_f32_16x16x4_f32(false, ones, false, b, (short)0, a5, false, false);
}

// ---------------------------------------------------------------------------
// Kernel 2: HBM-bound streaming column-sum of (in - tg) via WMMA row folding.
// Straight-line inner loop: one batched load clause, diffs, 6 WMMAs.
// ---------------------------------------------------------------------------
__global__ __launch_bounds__(256) void emd_colsum_wmma(const float* __restrict__ in,
                                                       const float* __restrict__ tg,
                                                       float* __restrict__ ws,
                                                       int nrows) {
    __shared__ float sws[96];

    const int lane          = threadIdx.x & 31;
    const int wavesPerBlock = blockDim.x >> 5;
    const int waveId        = blockIdx.x * wavesPerBlock + (threadIdx.x >> 5);
    const int numWaves      = gridDim.x * wavesPerBlock;

    const int n     = lane & 15;  // column within 16-wide tile (N across lanes)
    const int khalf = lane >> 4;  // lanes 0-15: k={0,1}; lanes 16-31: k={2,3}

    // Tile 5 (cols 80..95): clamp address to col 84, zero the value via mask.
    const float m5 = (n < 5) ? 1.0f : 0.0f;
    const int   c5 = (n < 5) ? (80 + n) : 84;

    if (threadIdx.x < 96) sws[threadIdx.x] = 0.0f;
    __syncthreads();

    v8f a0 = {}, a1 = {}, a2 = {}, a3 = {}, a4 = {}, a5 = {};
    v2f ones; ones[0] = 1.0f; ones[1] = 1.0f;

    const int nFull  = nrows & ~3;
    const int stride = numWaves * 4 * N_BINS;          // elements per grid-stride step
    const int laneA  = 2 * khalf * N_BINS + n;         // lane offset, tiles 0..4
    const int lane5  = 2 * khalf * N_BINS + c5;        // lane offset, tile 5 (clamped)

    const float* pInA = in + (size_t)waveId * 4 * N_BINS + laneA;
    const float* pTgA = tg + (size_t)waveId * 4 * N_BINS + laneA;
    const float* pIn5 = in + (size_t)waveId * 4 * N_BINS + lane5;
    const float* pTg5 = tg + (size_t)waveId * 4 * N_BINS + lane5;

#pragma unroll 2
    for (int r0 = waveId * 4; r0 < nFull; r0 += numWaves * 4) {
        float vi[12], vg[12];
        emd_load_group(pInA, pIn5, vi);
        emd_load_group(pTgA, pTg5, vg);
        emd_accum(vi, vg, m5, ones, a0, a1, a2, a3, a4, a5);
        pInA += stride;  pTgA += stride;
        pIn5 += stride;  pTg5 += stride;
    }

    // D VGPR0: lanes 0-15 hold (M=0, N=lane) = column sum for N=lane.
    if (lane < 16) {
        atomicAdd(&sws[ 0 + lane], a0[0]);
        atomicAdd(&sws[16 + lane], a1[0]);
        atomicAdd(&sws[32 + lane], a2[0]);
        atomicAdd(&sws[48 + lane], a3[0]);
        atomicAdd(&sws[64 + lane], a4[0]);
        if (lane < N_BINS - 80) atomicAdd(&sws[80 + lane], a5[0]);
    }
    __syncthreads();

    if (threadIdx.x < N_BINS) atomicAdd(&ws[threadIdx.x], sws[threadIdx.x]);
}

// ---------------------------------------------------------------------------
// Kernel 3: single wave. Adds tail rows (nrows % 4), takes |.|, applies the
// analytic harmonic weights W_j = sum_{m=j+1..85} 1/m, reduces, scales by 0.1.
// ---------------------------------------------------------------------------
__global__ __launch_bounds__(32) void emd_finalize(const float* __restrict__ in,
                                                   const float* __restrict__ tg,
                                                   const float* __restrict__ ws,
                                                   float* __restrict__ out,
                                                   int nrows) {
    const int lane  = threadIdx.x;
    const int nFull = nrows & ~3;

    float total = 0.0f;
    for (int c = lane; c < N_BINS; c += 32) {
        float s = ws[c];
        for (int r = nFull; r < nrows; ++r)          // tail rows (0..3)
            s += in[r * N_BINS + c] - tg[r * N_BINS + c];
        float d = fabsf(s);
        float w = 0.0f;
        for (int m = c + 1; m <= N_BINS; ++m) w += 1.0f / (float)m;
        total += d * w;
    }
#pragma unroll
    for (int off = 16; off > 0; off >>= 1)
        total += __shfl_down(total, off, 32);
    if (lane == 0) out[0] = total * 0.1f;
}

// ---------------------------------------------------------------------------
extern "C" void kernel_launch(void* const* d_in, const int* in_sizes, int n_in,
                              void* d_out, int out_size, void* d_ws, size_t ws_size,
                              hipStream_t stream) {
    const float* in = (const float*)d_in[0];
    const float* tg = (const float*)d_in[1];
    float* out = (float*)d_out;
    float* ws  = (float*)d_ws;

    const int nrows = in_sizes[0] / N_BINS;  // 500000

    emd_zero_ws<<<1, 128, 0, stream>>>(ws);
    emd_colsum_wmma<<<1024, 256, 0, stream>>>(in, tg, ws, nrows);
    emd_finalize<<<1, 32, 0, stream>>>(in, tg, ws, out, nrows);
}